// SimpleConv_70351564308901
// MI455X (gfx1250) — compile-verified
//
#include <hip/hip_runtime.h>

typedef __attribute__((ext_vector_type(2))) float v2f;
typedef __attribute__((ext_vector_type(8))) float v8f;

#define DIN   128
#define DOUT  48
#define WAVES_PER_BLOCK 8

// ---------------------------------------------------------------------------
// Kernel 1: h[N,48] = feat[N,128] @ W[128,48]  via V_WMMA_F32_16X16X4_F32.
// One wave -> one 16-row tile -> three 16x16 f32 accumulators (48 cols).
//
// A frag (16x4 f32, 2 VGPR): lane l<16: row l,    K = k0+{0,1}
//                            lane l>=16: row l-16, K = k0+2+{0,1}
//   -> contiguous float2 global load per lane.
// B frag (4x16 f32, 2 VGPR): VGPR v, lanes 0-15: K=k0+v, N=lane;
//                            lanes 16-31: K=k0+2+v, N=lane-16.
//   -> W staged in LDS as K-pairs: lW2[p][n] = {W[2p][n], W[2p+1][n]}
//      so each fragment is ONE aligned ds_load_b64 (no fix-up movs).
// C/D  (16x16 f32, 8 VGPR):  VGPR g: lanes 0-15 -> M=g, lanes 16-31 -> M=g+8.
// ---------------------------------------------------------------------------
__global__ __launch_bounds__(256) void gnn_gemm_wmma_f32(
    const float* __restrict__ feat, const float* __restrict__ W,
    float* __restrict__ h, int n_nodes) {
  __shared__ float2 lW2[(DIN / 2) * DOUT];  // 24 KB, K-pair swizzled
  for (int i = threadIdx.x; i < (DIN / 2) * DOUT; i += 256) {
    const int p = i / DOUT;        // K-pair index: rows 2p, 2p+1
    const int n = i - p * DOUT;    // output column
    lW2[i] = make_float2(W[(2 * p) * DOUT + n], W[(2 * p + 1) * DOUT + n]);
  }
  __syncthreads();

  const int lane = threadIdx.x & 31;
  const int wave = threadIdx.x >> 5;
  const int tile = blockIdx.x * WAVES_PER_BLOCK + wave;
  const int n_tiles = (n_nodes + 15) >> 4;
  if (tile >= n_tiles) return;

  const int row0 = tile << 4;
  const int hi   = lane >> 4;   // 0: lanes 0-15, 1: lanes 16-31
  const int lml  = lane & 15;

  int arow = row0 + lml;
  if (arow >= n_nodes) arow = n_nodes - 1;  // clamp (stores are guarded)
  const float* aptr = feat + (size_t)arow * DIN + 2 * hi;

  v8f acc0 = {}, acc1 = {}, acc2 = {};

#pragma unroll
  for (int k0 = 0; k0 < DIN; k0 += 4) {
    const float2 av = *(const float2*)(aptr + k0);   // K = k0+2*hi, +1
    v2f a; a.x = av.x; a.y = av.y;

    // pair index p = (k0 + 2*hi)/2 = k0/2 + hi
    const float2* wp = &lW2[((k0 >> 1) + hi) * DOUT + lml];
    const float2 f0 = wp[0];
    const float2 f1 = wp[16];
    const float2 f2 = wp[32];
    v2f b0, b1, b2;
    b0.x = f0.x; b0.y = f0.y;
    b1.x = f1.x; b1.y = f1.y;
    b2.x = f2.x; b2.y = f2.y;

    acc0 = __builtin_amdgcn_wmma_f32_16x16x4_f32(false, a, false, b0,
                                                 (short)0, acc0, false, false);
    acc1 = __builtin_amdgcn_wmma_f32_16x16x4_f32(false, a, false, b1,
                                                 (short)0, acc1, false, false);
    acc2 = __builtin_amdgcn_wmma_f32_16x16x4_f32(false, a, false, b2,
                                                 (short)0, acc2, false, false);
  }

#pragma unroll
  for (int g = 0; g < 8; ++g) {
    const int row = row0 + g + 8 * hi;
    if (row < n_nodes) {
      float* hp = h + (size_t)row * DOUT + lml;
      hp[0]  = acc0[g];
      hp[16] = acc1[g];
      hp[32] = acc2[g];
    }
  }
}

// ---------------------------------------------------------------------------
// Kernel 2: edge-weighted scatter-add.  12 threads per edge, 4 channels each:
//   out[dst, 4g..4g+3] += w_e * h[src, 4g..4g+3]
// h (19.2 MB) and out (19.2 MB) are both L2-resident (192 MB L2), so the
// ~600 MB of gather + atomic traffic never touches HBM.
// ---------------------------------------------------------------------------
__global__ __launch_bounds__(256) void gnn_edge_scatter(
    const float* __restrict__ h, const int* __restrict__ esrc,
    const int* __restrict__ edst, const float* __restrict__ ew,
    float* __restrict__ out, int n_edges) {
  const long long gid = (long long)blockIdx.x * 256 + threadIdx.x;
  const long long e = gid / 12;
  const int g = (int)(gid - e * 12);
  if (e >= n_edges) return;

  const int   s = esrc[e];
  const int   d = edst[e];
  const float w = ew[e];

  const float4 hv = *(const float4*)(h + (size_t)s * DOUT + 4 * g);
  float* op = out + (size_t)d * DOUT + 4 * g;
  atomicAdd(op + 0, w * hv.x);
  atomicAdd(op + 1, w * hv.y);
  atomicAdd(op + 2, w * hv.z);
  atomicAdd(op + 3, w * hv.w);
}

// ---------------------------------------------------------------------------
// Kernel 0: zero accumulator (harness poisons d_out).  Kernel 3: in-place ReLU.
// ---------------------------------------------------------------------------
__global__ __launch_bounds__(256) void gnn_zero(float4* __restrict__ p, int n4) {
  const int i = blockIdx.x * 256 + threadIdx.x;
  if (i < n4) p[i] = make_float4(0.f, 0.f, 0.f, 0.f);
}

__global__ __launch_bounds__(256) void gnn_relu(float4* __restrict__ p, int n4) {
  const int i = blockIdx.x * 256 + threadIdx.x;
  if (i < n4) {
    float4 v = p[i];
    v.x = fmaxf(v.x, 0.f); v.y = fmaxf(v.y, 0.f);
    v.z = fmaxf(v.z, 0.f); v.w = fmaxf(v.w, 0.f);
    p[i] = v;
  }
}

extern "C" void kernel_launch(void* const* d_in, const int* in_sizes, int n_in,
                              void* d_out, int out_size, void* d_ws, size_t ws_size,
                              hipStream_t stream) {
  const float* feat = (const float*)d_in[0];   // [N,128]
  const float* W    = (const float*)d_in[1];   // [128,48]
  const int*   esrc = (const int*)d_in[2];     // [E]
  const int*   edst = (const int*)d_in[3];     // [E]
  const float* ew   = (const float*)d_in[4];   // [E]
  float* out = (float*)d_out;                  // [N,48]
  float* h   = (float*)d_ws;                   // scratch: N*48 floats (19.2 MB)

  const int n_nodes = in_sizes[0] / DIN;
  const int n_edges = in_sizes[2];

  // 0) zero the accumulator
  const int n4 = (n_nodes * DOUT) / 4;
  gnn_zero<<<(n4 + 255) / 256, 256, 0, stream>>>((float4*)out, n4);

  // 1) dense projection via f32 WMMA
  const int n_tiles = (n_nodes + 15) / 16;
  const int gemm_blocks = (n_tiles + WAVES_PER_BLOCK - 1) / WAVES_PER_BLOCK;
  gnn_gemm_wmma_f32<<<gemm_blocks, 256, 0, stream>>>(feat, W, h, n_nodes);

  // 2) edge-weighted scatter-add (12 threads per edge)
  const long long nthreads = (long long)n_edges * 12;
  const int sc_blocks = (int)((nthreads + 255) / 256);
  gnn_edge_scatter<<<sc_blocks, 256, 0, stream>>>(h, esrc, edst, ew, out, n_edges);

  // 3) ReLU epilogue in place
  gnn_relu<<<(n4 + 255) / 256, 256, 0, stream>>>((float4*)out, n4);
}